// NoisyRouter_74569222193396
// MI455X (gfx1250) — compile-verified
//
#include <hip/hip_runtime.h>
#include <hip/hip_bf16.h>

typedef float v2f __attribute__((ext_vector_type(2)));
typedef float v8f __attribute__((ext_vector_type(8)));

namespace {
constexpr int Dk    = 4096;      // reduction dim
constexpr int Ek    = 64;        // experts
constexpr int KC    = 128;       // K chunk staged in LDS
constexpr int P1    = KC + 4;    // WrT pitch (floats): 132 -> conflict-free b64 reads
constexpr int P2    = 68;        // logits pitch (floats)
constexpr int NW    = 8;         // waves per block
constexpr int TPB   = NW * 16;   // tokens per block = 128
constexpr int K_TOP = 8;
}

__global__ __launch_bounds__(256, 1)
void router_wmma_kernel(const float* __restrict__ x,
                        const float* __restrict__ Wr,
                        const float* __restrict__ br,
                        float* __restrict__ scores,
                        int* __restrict__ topk_idx)
{
    // max(P1*Ek, P2*TPB) = max(8448, 8704) floats
    __shared__ __align__(16) float smem[P2 * TPB];
    float* WrT = smem;

    const int tid  = threadIdx.x;
    const int lane = tid & 31;
    const int wave = tid >> 5;
    const int lm   = lane & 15;   // M row (A) / N col (B,C) within tile
    const int hi   = lane >> 4;   // lane half -> K offset pairing

    const int tokBase = blockIdx.x * TPB + wave * 16;
    const float* xa = x + (size_t)(tokBase + lm) * Dk + 2 * hi;

    v8f acc0 = {}, acc1 = {}, acc2 = {}, acc3 = {};

    for (int kc = 0; kc < Dk; kc += KC) {
        // Cooperative transposed stage of Wr[kc:kc+KC][0:64] -> WrT[e*P1 + k]
        {
            const int r  = tid & (KC - 1);   // row within chunk (128 rows, 256 thr)
            const int eh = tid >> 7;         // which 32-expert half
            const float* wrow = Wr + (size_t)(kc + r) * Ek + eh * 32;
            #pragma unroll
            for (int e4 = 0; e4 < 8; ++e4) {
                float4 q = *(const float4*)(wrow + e4 * 4);
                int eb = eh * 32 + e4 * 4;
                WrT[(size_t)(eb + 0) * P1 + r] = q.x;
                WrT[(size_t)(eb + 1) * P1 + r] = q.y;
                WrT[(size_t)(eb + 2) * P1 + r] = q.z;
                WrT[(size_t)(eb + 3) * P1 + r] = q.w;
            }
        }
        __syncthreads();

        const float* xk = xa + kc;
        #pragma unroll
        for (int k = 0; k < KC; k += 4) {
            // A 16x4 f32 fragment: lane lm row, K = k + 2*hi (+1)
            v2f a = *(const v2f*)(xk + k);
            // B 4x16 fragments: column-major in LDS -> single b64 per tile
            const float* bp = &WrT[k + 2 * hi];
            v2f b0 = *(const v2f*)(bp + (size_t)(lm +  0) * P1);
            v2f b1 = *(const v2f*)(bp + (size_t)(lm + 16) * P1);
            v2f b2 = *(const v2f*)(bp + (size_t)(lm + 32) * P1);
            v2f b3 = *(const v2f*)(bp + (size_t)(lm + 48) * P1);
            acc0 = __builtin_amdgcn_wmma_f32_16x16x4_f32(false, a, false, b0, (short)0, acc0, false, false);
            acc1 = __builtin_amdgcn_wmma_f32_16x16x4_f32(false, a, false, b1, (short)0, acc1, false, false);
            acc2 = __builtin_amdgcn_wmma_f32_16x16x4_f32(false, a, false, b2, (short)0, acc2, false, false);
            acc3 = __builtin_amdgcn_wmma_f32_16x16x4_f32(false, a, false, b3, (short)0, acc3, false, false);
        }
        __syncthreads();
    }

    // Router bias: each lane's fragment column is one expert (N = lm + 16*t)
    float bb0 = br[lm], bb1 = br[lm + 16], bb2 = br[lm + 32], bb3 = br[lm + 48];
    #pragma unroll
    for (int r = 0; r < 8; ++r) {
        acc0[r] += bb0; acc1[r] += bb1; acc2[r] += bb2; acc3[r] += bb3;
    }

    // Spill logits tile to LDS rows by token: C vgpr r, lane half hi -> M = r + 8*hi
    float* LG = smem;
    #pragma unroll
    for (int r = 0; r < 8; ++r) {
        int m = wave * 16 + r + 8 * hi;
        LG[(size_t)m * P2 + ( 0 + lm)] = acc0[r];
        LG[(size_t)m * P2 + (16 + lm)] = acc1[r];
        LG[(size_t)m * P2 + (32 + lm)] = acc2[r];
        LG[(size_t)m * P2 + (48 + lm)] = acc3[r];
    }
    __syncthreads();

    // Top-8 + sparse softmax: one lane per token
    if (lane < 16) {
        const int tok = tokBase + lane;
        const float* row = &LG[(size_t)(wave * 16 + lane) * P2];
        float v[64];
        #pragma unroll
        for (int e4 = 0; e4 < 16; ++e4) {
            float4 q = *(const float4*)(row + e4 * 4);
            v[e4 * 4 + 0] = q.x; v[e4 * 4 + 1] = q.y;
            v[e4 * 4 + 2] = q.z; v[e4 * 4 + 3] = q.w;
        }
        float bv[K_TOP]; int bi[K_TOP];
        #pragma unroll
        for (int p = 0; p < K_TOP; ++p) {
            float m = v[0]; int mi = 0;
            #pragma unroll
            for (int e = 1; e < 64; ++e) {
                if (v[e] > m) { m = v[e]; mi = e; }   // strict > : lowest idx on ties
            }
            bv[p] = m; bi[p] = mi;
            #pragma unroll
            for (int e = 0; e < 64; ++e) v[e] = (e == mi) ? -__builtin_inff() : v[e];
        }
        // softmax over the kept 8 (all others are exp(-inf) = 0)
        float mx = bv[0], s = 0.f, w[K_TOP];
        #pragma unroll
        for (int p = 0; p < K_TOP; ++p) { w[p] = __expf(bv[p] - mx); s += w[p]; }
        float inv = 1.0f / s;
        #pragma unroll
        for (int p = 0; p < K_TOP; ++p) w[p] *= inv;

        // dense 64-wide score row (zeros except top-8), built with selects
        float* orow = scores + (size_t)tok * Ek;
        #pragma unroll
        for (int e4 = 0; e4 < 16; ++e4) {
            float o0 = 0.f, o1 = 0.f, o2 = 0.f, o3 = 0.f;
            #pragma unroll
            for (int p = 0; p < K_TOP; ++p) {
                o0 = (bi[p] == e4 * 4 + 0) ? w[p] : o0;
                o1 = (bi[p] == e4 * 4 + 1) ? w[p] : o1;
                o2 = (bi[p] == e4 * 4 + 2) ? w[p] : o2;
                o3 = (bi[p] == e4 * 4 + 3) ? w[p] : o3;
            }
            float4 q; q.x = o0; q.y = o1; q.z = o2; q.w = o3;
            *(float4*)(orow + e4 * 4) = q;
        }
        int* irow = topk_idx + (size_t)tok * K_TOP;
        int4 i0 = make_int4(bi[0], bi[1], bi[2], bi[3]);
        int4 i1 = make_int4(bi[4], bi[5], bi[6], bi[7]);
        *(int4*)(irow + 0) = i0;
        *(int4*)(irow + 4) = i1;
    }
}

extern "C" void kernel_launch(void* const* d_in, const int* in_sizes, int n_in,
                              void* d_out, int out_size, void* d_ws, size_t ws_size,
                              hipStream_t stream) {
    const float* x  = (const float*)d_in[0];
    const float* Wr = (const float*)d_in[1];
    const float* br = (const float*)d_in[2];
    // d_in[3] = Wn, d_in[4] = bn feed a provably dead branch in the reference
    // (noisy_vec is computed then deleted; top-k/softmax use clean logits) -> skipped.
    (void)n_in; (void)d_ws; (void)ws_size; (void)out_size;

    const int T = in_sizes[0] / Dk;          // 16384 tokens (B*S)
    float* scores = (float*)d_out;           // [T, 64] f32
    int*   idx    = (int*)(scores + (size_t)T * Ek);  // [T, 8] i32, concatenated

    dim3 grid(T / TPB), block(256);
    hipLaunchKernelGGL(router_wmma_kernel, grid, block, 0, stream,
                       x, Wr, br, scores, idx);
}